// RecurrentTransformer_18305150615780
// MI455X (gfx1250) — compile-verified
//
#include <hip/hip_runtime.h>
#include <math.h>

// ---------------- problem constants ----------------
#define BB   1024
#define SS   50
#define DD   96
#define HH   8
#define DH   12
#define DFF  192
#define LL   3
#define NCYC 3
#define VV   40000

typedef __attribute__((ext_vector_type(2))) float v2f;
typedef __attribute__((ext_vector_type(8))) float v8f;

__device__ __forceinline__ float gelu_f(float x) {
    // exact GELU (approximate=False): 0.5*x*(1+erf(x/sqrt(2)))
    return 0.5f * x * (1.0f + erff(x * 0.7071067811865475f));
}
__device__ __forceinline__ float sigmoid_f(float x) {
    return 1.0f / (1.0f + expf(-x));
}

// =====================================================================
// Generic GEMM via V_WMMA_F32_16X16X4_F32 (wave32, 16x16 tiles).
//   C[M,N] = act(A[M,K] @ W[N,K]^T + bias[N])     (ldc allows strided/offset C)
// Each wave computes a 32(M) x 32(N) tile: 4 accumulators; both A fragments
// are reused across both N-tiles and both B fragments across both M-tiles,
// so the k-step is 4 loads : 4 WMMAs (vs 3:2 for a 16x32 strip).
// Fragment layouts per CDNA5 ISA 7.12.2:
//   A 16x4 f32 : lane L (L<16): M=L, VGPR0=K0,VGPR1=K1 ; lanes 16-31: K2/K3
//   B 4x16 f32 : mirrored (lane = N + 16*(K>=2), VGPR = K&1)
//   C 16x16 f32: VGPR r, lane L -> M = r + 8*(L>=16), N = L&15
// ACT: 0 = none, 1 = gelu, 2 = sigmoid
// Requires M%32==0, N%32==0, K%4==0 (true for every GEMM in this model).
// =====================================================================
template <int ACT>
__global__ void gemm_wmma(const float* __restrict__ A, const float* __restrict__ W,
                          const float* __restrict__ bias, float* __restrict__ C,
                          int M, int N, int K, int ldc)
{
    const int lane   = threadIdx.x & 31;
    const int wave   = (int)((blockIdx.x * blockDim.x + threadIdx.x) >> 5);
    const int ntiles = N >> 5;                 // 32 columns per wave
    const int total  = (M >> 5) * ntiles;
    if (wave >= total) return;                 // wave-uniform branch (EXEC stays all-1)

    const int nt = wave % ntiles;
    const int mt = wave / ntiles;
    const int m0 = mt << 5, n0 = nt << 5;
    const int half = lane >> 4;                // 0: K{0,1}, 1: K{2,3}
    const int lm   = lane & 15;
    const int kb   = half << 1;

    const float* A0 = A + (size_t)(m0 + lm) * K + kb;
    const float* A1 = A + (size_t)(m0 + 16 + lm) * K + kb;
    const float* W0 = W + (size_t)(n0 + lm) * K + kb;
    const float* W1 = W + (size_t)(n0 + 16 + lm) * K + kb;

    const float b0v = bias[n0 + lm];
    const float b1v = bias[n0 + 16 + lm];
    v8f acc00, acc01, acc10, acc11;            // acc[mtile][ntile]
#pragma unroll
    for (int i = 0; i < 8; ++i) {
        acc00[i] = b0v; acc01[i] = b1v;
        acc10[i] = b0v; acc11[i] = b1v;
    }

    for (int k0 = 0; k0 < K; k0 += 4) {
        v2f a0 = *(const v2f*)(A0 + k0);
        v2f a1 = *(const v2f*)(A1 + k0);
        v2f f0 = *(const v2f*)(W0 + k0);
        v2f f1 = *(const v2f*)(W1 + k0);
        acc00 = __builtin_amdgcn_wmma_f32_16x16x4_f32(false, a0, false, f0, (short)0, acc00, false, false);
        acc01 = __builtin_amdgcn_wmma_f32_16x16x4_f32(false, a0, false, f1, (short)0, acc01, false, false);
        acc10 = __builtin_amdgcn_wmma_f32_16x16x4_f32(false, a1, false, f0, (short)0, acc10, false, false);
        acc11 = __builtin_amdgcn_wmma_f32_16x16x4_f32(false, a1, false, f1, (short)0, acc11, false, false);
    }

#pragma unroll
    for (int r = 0; r < 8; ++r) {
        const int row0 = m0 + (half << 3) + r;
        const int row1 = row0 + 16;
        float y00 = acc00[r], y01 = acc01[r], y10 = acc10[r], y11 = acc11[r];
        if (ACT == 1) { y00 = gelu_f(y00); y01 = gelu_f(y01); y10 = gelu_f(y10); y11 = gelu_f(y11); }
        if (ACT == 2) { y00 = sigmoid_f(y00); y01 = sigmoid_f(y01); y10 = sigmoid_f(y10); y11 = sigmoid_f(y11); }
        C[(size_t)row0 * ldc + n0 + lm]      = y00;
        C[(size_t)row0 * ldc + n0 + 16 + lm] = y01;
        C[(size_t)row1 * ldc + n0 + lm]      = y10;
        C[(size_t)row1 * ldc + n0 + 16 + lm] = y11;
    }
}

// =====================================================================
// Input embedding: x[b,s,:] = loc_emb[loc] + user_emb[users[b,0]] +
//                  concat(hour_emb, wd_emb) @ temp_w^T + temp_b + pos[s]
// one block (96 threads) per token
// =====================================================================
__global__ void embed_kernel(const int* __restrict__ loc, const int* __restrict__ usr,
                             const int* __restrict__ smin, const int* __restrict__ wd,
                             const float* __restrict__ loc_emb, const float* __restrict__ user_emb,
                             const float* __restrict__ hour_emb, const float* __restrict__ wd_emb,
                             const float* __restrict__ temp_w, const float* __restrict__ temp_b,
                             const float* __restrict__ pos, float* __restrict__ x)
{
    const int bs = blockIdx.x;             // 0 .. B*S-1
    const int b  = bs / SS, s = bs % SS;
    const int c  = threadIdx.x;            // 0 .. 95
    __shared__ float t[48];
    if (c < 24) {
        int h = (smin[bs] / 60) % 24;
        t[c] = hour_emb[h * 24 + c];
    } else if (c < 48) {
        t[c] = wd_emb[wd[bs] * 24 + (c - 24)];
    }
    __syncthreads();
    float acc = temp_b[c];
#pragma unroll
    for (int j = 0; j < 48; ++j) acc = fmaf(t[j], temp_w[c * 48 + j], acc);
    acc += loc_emb[(size_t)loc[bs] * DD + c];
    acc += user_emb[(size_t)usr[b * SS] * DD + c];
    acc += pos[s * DD + c];
    x[(size_t)bs * DD + c] = acc;
}

// =====================================================================
// Self-attention over tokens (dh = 12, S = 50): one thread per (b,s,h).
// qkv row layout per token: [q(96) | k(96) | v(96)]
// =====================================================================
__global__ void attn_self(const float* __restrict__ qkv, float* __restrict__ o)
{
    const int idx = blockIdx.x * blockDim.x + threadIdx.x;
    if (idx >= BB * SS * HH) return;
    const int h  = idx % HH;
    const int bs = idx / HH;
    const int b  = bs / SS;
    const float scale = 0.2886751345948129f; // 1/sqrt(12)

    const float* qr = qkv + (size_t)bs * (3 * DD) + h * DH;
    float q[DH];
#pragma unroll
    for (int d = 0; d < DH; ++d) q[d] = qr[d] * scale;

    float sc[SS];
    float mx = -1e30f;
    for (int j = 0; j < SS; ++j) {
        const float* kr = qkv + (size_t)(b * SS + j) * (3 * DD) + DD + h * DH;
        float dot = 0.f;
#pragma unroll
        for (int d = 0; d < DH; ++d) dot = fmaf(q[d], kr[d], dot);
        sc[j] = dot;
        mx = fmaxf(mx, dot);
    }
    float sum = 0.f;
    for (int j = 0; j < SS; ++j) { sc[j] = expf(sc[j] - mx); sum += sc[j]; }
    const float inv = 1.f / sum;

    float out[DH];
#pragma unroll
    for (int d = 0; d < DH; ++d) out[d] = 0.f;
    for (int j = 0; j < SS; ++j) {
        const float w = sc[j] * inv;
        const float* vr = qkv + (size_t)(b * SS + j) * (3 * DD) + 2 * DD + h * DH;
#pragma unroll
        for (int d = 0; d < DH; ++d) out[d] = fmaf(w, vr[d], out[d]);
    }
    float* orow = o + (size_t)bs * DD + h * DH;
#pragma unroll
    for (int d = 0; d < DH; ++d) orow[d] = out[d];
}

// =====================================================================
// Cross-attention: 1 query (hidden) over S precomputed K/V rows.
// kv row layout per token: [k(96) | v(96)]. one thread per (b,h).
// =====================================================================
__global__ void attn_cross(const float* __restrict__ qh, const float* __restrict__ kv,
                           float* __restrict__ o)
{
    const int idx = blockIdx.x * blockDim.x + threadIdx.x;
    if (idx >= BB * HH) return;
    const int h = idx % HH;
    const int b = idx / HH;
    const float scale = 0.2886751345948129f;

    const float* qr = qh + (size_t)b * DD + h * DH;
    float q[DH];
#pragma unroll
    for (int d = 0; d < DH; ++d) q[d] = qr[d] * scale;

    float sc[SS];
    float mx = -1e30f;
    for (int j = 0; j < SS; ++j) {
        const float* kr = kv + (size_t)(b * SS + j) * (2 * DD) + h * DH;
        float dot = 0.f;
#pragma unroll
        for (int d = 0; d < DH; ++d) dot = fmaf(q[d], kr[d], dot);
        sc[j] = dot;
        mx = fmaxf(mx, dot);
    }
    float sum = 0.f;
    for (int j = 0; j < SS; ++j) { sc[j] = expf(sc[j] - mx); sum += sc[j]; }
    const float inv = 1.f / sum;

    float out[DH];
#pragma unroll
    for (int d = 0; d < DH; ++d) out[d] = 0.f;
    for (int j = 0; j < SS; ++j) {
        const float w = sc[j] * inv;
        const float* vr = kv + (size_t)(b * SS + j) * (2 * DD) + DD + h * DH;
#pragma unroll
        for (int d = 0; d < DH; ++d) out[d] = fmaf(w, vr[d], out[d]);
    }
    float* orow = o + (size_t)b * DD + h * DH;
#pragma unroll
    for (int d = 0; d < DH; ++d) orow[d] = out[d];
}

// =====================================================================
// out = LN(a + b) * g + beta  over D=96, one wave32 per row, shfl reduce
// =====================================================================
__global__ void add_ln(const float* __restrict__ a, const float* __restrict__ b,
                       const float* __restrict__ g, const float* __restrict__ be,
                       float* __restrict__ out, int rows)
{
    const int row  = (int)((blockIdx.x * blockDim.x + threadIdx.x) >> 5);
    const int lane = threadIdx.x & 31;
    if (row >= rows) return;
    const float* ar = a + (size_t)row * DD;
    const float* br = b + (size_t)row * DD;
    float v[3];
    float s = 0.f;
#pragma unroll
    for (int i = 0; i < 3; ++i) { v[i] = ar[lane + 32 * i] + br[lane + 32 * i]; s += v[i]; }
#pragma unroll
    for (int off = 16; off >= 1; off >>= 1) s += __shfl_xor(s, off, 32);
    const float mean = s * (1.0f / 96.0f);
    float qv = 0.f;
#pragma unroll
    for (int i = 0; i < 3; ++i) { float d = v[i] - mean; qv += d * d; }
#pragma unroll
    for (int off = 16; off >= 1; off >>= 1) qv += __shfl_xor(qv, off, 32);
    const float rs = rsqrtf(qv * (1.0f / 96.0f) + 1e-5f);
    float* orow = out + (size_t)row * DD;
#pragma unroll
    for (int i = 0; i < 3; ++i) {
        int j = lane + 32 * i;
        orow[j] = (v[i] - mean) * rs * g[j] + be[j];
    }
}

// gated = LN(g*c1 + (1-g)*c2 + hidden), cc row layout = [c1(96)|c2(96)]
__global__ void gate_ln(const float* __restrict__ gbuf, const float* __restrict__ cc,
                        const float* __restrict__ hid, const float* __restrict__ g,
                        const float* __restrict__ be, float* __restrict__ out)
{
    const int row  = (int)((blockIdx.x * blockDim.x + threadIdx.x) >> 5);
    const int lane = threadIdx.x & 31;
    if (row >= BB) return;
    const float* gr = gbuf + (size_t)row * DD;
    const float* c1 = cc + (size_t)row * 2 * DD;
    const float* c2 = c1 + DD;
    const float* hr = hid + (size_t)row * DD;
    float v[3];
    float s = 0.f;
#pragma unroll
    for (int i = 0; i < 3; ++i) {
        int j = lane + 32 * i;
        float gg = gr[j];
        v[i] = gg * c1[j] + (1.f - gg) * c2[j] + hr[j];
        s += v[i];
    }
#pragma unroll
    for (int off = 16; off >= 1; off >>= 1) s += __shfl_xor(s, off, 32);
    const float mean = s * (1.0f / 96.0f);
    float qv = 0.f;
#pragma unroll
    for (int i = 0; i < 3; ++i) { float d = v[i] - mean; qv += d * d; }
#pragma unroll
    for (int off = 16; off >= 1; off >>= 1) qv += __shfl_xor(qv, off, 32);
    const float rs = rsqrtf(qv * (1.0f / 96.0f) + 1e-5f);
    float* orow = out + (size_t)row * DD;
#pragma unroll
    for (int i = 0; i < 3; ++i) {
        int j = lane + 32 * i;
        orow[j] = (v[i] - mean) * rs * g[j] + be[j];
    }
}

__global__ void bcast_hidden(const float* __restrict__ init, float* __restrict__ hidden)
{
    int i = blockIdx.x * blockDim.x + threadIdx.x;
    if (i < BB * DD) hidden[i] = init[i % DD];
}

// ---------------------------------------------------------------------
static inline void launch_gemm(int act, const float* A, const float* W, const float* bias,
                               float* C, int M, int N, int K, int ldc, hipStream_t stream)
{
    int waves  = (M >> 5) * (N >> 5);          // 32x32 tile per wave
    int blocks = (waves * 32 + 255) / 256;
    switch (act) {
        case 1:  gemm_wmma<1><<<blocks, 256, 0, stream>>>(A, W, bias, C, M, N, K, ldc); break;
        case 2:  gemm_wmma<2><<<blocks, 256, 0, stream>>>(A, W, bias, C, M, N, K, ldc); break;
        default: gemm_wmma<0><<<blocks, 256, 0, stream>>>(A, W, bias, C, M, N, K, ldc); break;
    }
}

extern "C" void kernel_launch(void* const* d_in, const int* in_sizes, int n_in,
                              void* d_out, int out_size, void* d_ws, size_t ws_size,
                              hipStream_t stream)
{
    (void)in_sizes; (void)n_in; (void)out_size; (void)ws_size;

    const int*   locations  = (const int*)d_in[0];
    const int*   users      = (const int*)d_in[1];
    const int*   start_mins = (const int*)d_in[2];
    const int*   weekdays   = (const int*)d_in[3];
    const float* loc_emb    = (const float*)d_in[4];
    const float* user_emb   = (const float*)d_in[5];
    const float* hour_emb   = (const float*)d_in[6];
    const float* wd_emb     = (const float*)d_in[7];
    const float* temp_w     = (const float*)d_in[8];
    const float* temp_b     = (const float*)d_in[9];
    const float* pos_embed  = (const float*)d_in[10];
    const float* init_hidden= (const float*)d_in[11];
    const float* out_w1     = (const float*)d_in[12];
    const float* out_b1     = (const float*)d_in[13];
    const float* out_w2     = (const float*)d_in[14];
    const float* out_b2     = (const float*)d_in[15];
    const float* sa_wi = (const float*)d_in[16];
    const float* sa_bi = (const float*)d_in[17];
    const float* sa_wo = (const float*)d_in[18];
    const float* sa_bo = (const float*)d_in[19];
    const float* ci_wi = (const float*)d_in[20];
    const float* ci_bi = (const float*)d_in[21];
    const float* ci_wo = (const float*)d_in[22];
    const float* ci_bo = (const float*)d_in[23];
    const float* ch_wi = (const float*)d_in[24];
    const float* ch_bi = (const float*)d_in[25];
    const float* ch_wo = (const float*)d_in[26];
    const float* ch_bo = (const float*)d_in[27];
    const float* ffn_w1 = (const float*)d_in[28];
    const float* ffn_b1 = (const float*)d_in[29];
    const float* ffn_w2 = (const float*)d_in[30];
    const float* ffn_b2 = (const float*)d_in[31];
    const float* gate_w = (const float*)d_in[32];
    const float* gate_b = (const float*)d_in[33];
    const float* n1_g = (const float*)d_in[34];
    const float* n1_b = (const float*)d_in[35];
    const float* n2_g = (const float*)d_in[36];
    const float* n2_b = (const float*)d_in[37];
    const float* n3_g = (const float*)d_in[38];
    const float* n3_b = (const float*)d_in[39];

    // ---------------- workspace layout (floats) ----------------
    float* ws = (float*)d_ws;
    size_t off = 0;
    float* xbuf   = ws + off; off += (size_t)BB * SS * DD;           // 4.9M
    float* qkvbuf = ws + off; off += (size_t)BB * SS * 3 * DD;       // 14.7M
    float* t1     = ws + off; off += (size_t)BB * SS * DD;           // 4.9M
    float* kvbuf  = ws + off; off += (size_t)3 * BB * SS * 2 * DD;   // 29.5M (per-layer K/V)
    float* hidden = ws + off; off += (size_t)BB * DD;
    float* qh     = ws + off; off += (size_t)BB * DD;
    float* attn1  = ws + off; off += (size_t)BB * DD;
    float* vch    = ws + off; off += (size_t)BB * DD;
    float* cc     = ws + off; off += (size_t)BB * 2 * DD;            // [c1|c2]
    float* gbuf   = ws + off; off += (size_t)BB * DD;
    float* gated  = ws + off; off += (size_t)BB * DD;
    float* f1     = ws + off; off += (size_t)BB * DFF;
    float* f2     = ws + off; off += (size_t)BB * DD;
    float* h1     = ws + off; off += (size_t)BB * DD;
    float* t2     = qkvbuf;  // alias: qkv dead once attn_self consumed it

    // 1) embeddings -> x
    embed_kernel<<<BB * SS, 96, 0, stream>>>(locations, users, start_mins, weekdays,
                                             loc_emb, user_emb, hour_emb, wd_emb,
                                             temp_w, temp_b, pos_embed, xbuf);

    // 2) per-layer precompute (x never changes inside cycles):
    //    a1 = LN(selfattn(x)+x); kv[l] = a1 @ ci_wi[96:288]^T + bias
    const int rowsBS = BB * SS;
    for (int l = 0; l < LL; ++l) {
        launch_gemm(0, xbuf, sa_wi + (size_t)l * 3 * DD * DD, sa_bi + l * 3 * DD,
                    qkvbuf, rowsBS, 3 * DD, DD, 3 * DD, stream);
        int n = rowsBS * HH;
        attn_self<<<(n + 255) / 256, 256, 0, stream>>>(qkvbuf, t1);
        launch_gemm(0, t1, sa_wo + (size_t)l * DD * DD, sa_bo + l * DD,
                    t2, rowsBS, DD, DD, DD, stream);
        add_ln<<<(rowsBS * 32 + 255) / 256, 256, 0, stream>>>(t2, xbuf,
                    n1_g + l * DD, n1_b + l * DD, t1, rowsBS);
        launch_gemm(0, t1, ci_wi + (size_t)l * 3 * DD * DD + (size_t)DD * DD,
                    ci_bi + l * 3 * DD + DD,
                    kvbuf + (size_t)l * rowsBS * 2 * DD, rowsBS, 2 * DD, DD, 2 * DD, stream);
    }

    // 3) hidden = broadcast(init_hidden)
    bcast_hidden<<<(BB * DD + 255) / 256, 256, 0, stream>>>(init_hidden, hidden);

    // 4) recurrent cell: NC cycles x L layers
    for (int cyc = 0; cyc < NCYC; ++cyc) {
        for (int l = 0; l < LL; ++l) {
            const float* kvl = kvbuf + (size_t)l * rowsBS * 2 * DD;
            // c1 = crossattn(hidden -> a1 K/V) @ ci_wo^T + ci_bo
            launch_gemm(0, hidden, ci_wi + (size_t)l * 3 * DD * DD, ci_bi + l * 3 * DD,
                        qh, BB, DD, DD, DD, stream);
            attn_cross<<<(BB * HH + 255) / 256, 256, 0, stream>>>(qh, kvl, attn1);
            launch_gemm(0, attn1, ci_wo + (size_t)l * DD * DD, ci_bo + l * DD,
                        cc, BB, DD, DD, 2 * DD, stream);
            // c2 = (hidden @ ch_wi[192:288]^T + b) @ ch_wo^T + bo  (single-key softmax == 1)
            launch_gemm(0, hidden, ch_wi + (size_t)l * 3 * DD * DD + (size_t)2 * DD * DD,
                        ch_bi + l * 3 * DD + 2 * DD, vch, BB, DD, DD, DD, stream);
            launch_gemm(0, vch, ch_wo + (size_t)l * DD * DD, ch_bo + l * DD,
                        cc + DD, BB, DD, DD, 2 * DD, stream);
            // g = sigmoid([c1|c2] @ gate_w^T + gate_b)
            launch_gemm(2, cc, gate_w + (size_t)l * DD * 2 * DD, gate_b + l * DD,
                        gbuf, BB, DD, 2 * DD, DD, stream);
            // gated = LN(g*c1 + (1-g)*c2 + hidden)
            gate_ln<<<(BB * 32 + 255) / 256, 256, 0, stream>>>(gbuf, cc, hidden,
                        n2_g + l * DD, n2_b + l * DD, gated);
            // FFN + LN -> hidden
            launch_gemm(1, gated, ffn_w1 + (size_t)l * DFF * DD, ffn_b1 + l * DFF,
                        f1, BB, DFF, DD, DFF, stream);
            launch_gemm(0, f1, ffn_w2 + (size_t)l * DD * DFF, ffn_b2 + l * DD,
                        f2, BB, DD, DFF, DD, stream);
            add_ln<<<(BB * 32 + 255) / 256, 256, 0, stream>>>(f2, gated,
                        n3_g + l * DD, n3_b + l * DD, hidden, BB);
        }
    }

    // 5) logits = gelu(hidden @ out_w1^T + b1) @ out_w2^T + b2   (164 MB write; BW-bound)
    launch_gemm(1, hidden, out_w1, out_b1, h1, BB, DD, DD, DD, stream);
    launch_gemm(0, h1, out_w2, out_b2, (float*)d_out, BB, VV, DD, VV, stream);
}